// MoleculeGNN_20323785245081
// MI455X (gfx1250) — compile-verified
//
#include <hip/hip_runtime.h>

// ---------------------------------------------------------------------------
// MoleculeGNN for MI455X (gfx1250, wave32, WMMA).
//  * GEMMs use V_WMMA_F32_16X16X4_F32; one wave computes a 16x32 strip,
//    all fragments are contiguous b64 loads (weights pre-transposed).
//  * Edge aggregation is CSR-gather: per-launch counting sort of edges by dst
//    (~1.2M int atomics) then one wave per node accumulates 128 channels in
//    registers (float4/lane x 32 lanes) -- no f32 atomics on the hot path.
//  * Whole working set ~60MB < 192MB L2 on MI455X -> L2-bandwidth bound.
// ---------------------------------------------------------------------------

#define NN     50000      // nodes
#define EE     600000     // edges
#define FIN    9          // atom features
#define HH     128        // hidden dim
#define GG     2048       // graphs
#define BN_EPS 1e-5f

typedef __attribute__((ext_vector_type(2))) float v2f;
typedef __attribute__((ext_vector_type(8))) float v8f;

// ---------------------------------------------------------------- utilities
__global__ void fill_f32(float* __restrict__ p, float v, int n) {
    int i = blockIdx.x * blockDim.x + threadIdx.x;
    if (i < n) p[i] = v;
}

__global__ void fill_i32(int* __restrict__ p, int v, int n) {
    int i = blockIdx.x * blockDim.x + threadIdx.x;
    if (i < n) p[i] = v;
}

__global__ void copy_i32(const int* __restrict__ a, int* __restrict__ b, int n) {
    int i = blockIdx.x * blockDim.x + threadIdx.x;
    if (i < n) b[i] = a[i];
}

// ------------------------------------------------------------- CSR building
// counts[dst[e]]++ (counts pre-zeroed)
__global__ void hist_kernel(const int* __restrict__ dst, int* __restrict__ counts, int e) {
    int i = blockIdx.x * blockDim.x + threadIdx.x;
    if (i < e) atomicAdd(&counts[dst[i]], 1);
}

// single-block (1024 threads) in-place exclusive scan; data[n] = total.
__global__ void scan_kernel(int* __restrict__ data, int n) {
    __shared__ int part[1024];
    int t = threadIdx.x;
    int chunk = (n + 1023) / 1024;
    int base = t * chunk;
    int s = 0;
    for (int i = 0; i < chunk; ++i) {
        int j = base + i;
        if (j < n) s += data[j];
    }
    part[t] = s;
    __syncthreads();
    if (t == 0) {
        int run = 0;
        for (int i = 0; i < 1024; ++i) { int v = part[i]; part[i] = run; run += v; }
        data[n] = run;      // total == E
    }
    __syncthreads();
    int run = part[t];
    for (int i = 0; i < chunk; ++i) {
        int j = base + i;
        if (j < n) { int v = data[j]; data[j] = run; run += v; }
    }
}

// dis[n] = rsqrt(deg + 1), deg from CSR row pointers
__global__ void dis_kernel(const int* __restrict__ rowstart, float* __restrict__ dis, int n) {
    int i = blockIdx.x * blockDim.x + threadIdx.x;
    if (i < n) dis[i] = rsqrtf((float)(rowstart[i + 1] - rowstart[i]) + 1.0f);
}

// eidx[p] = e, bucketed by dst via cursor (copy of rowstart)
__global__ void scatter_kernel(const int* __restrict__ dst, int* __restrict__ cursor,
                               int* __restrict__ eidx, int e) {
    int i = blockIdx.x * blockDim.x + threadIdx.x;
    if (i < e) {
        int p = atomicAdd(&cursor[dst[i]], 1);
        eidx[p] = i;
    }
}

// ------------------------------------------------------- layer-0 tiny GEMM
__global__ void gemm0_kernel(const float* __restrict__ x, const float* __restrict__ W,
                             float* __restrict__ out) {
    int idx = blockIdx.x * 256 + threadIdx.x;   // exactly N*H threads
    int n = idx >> 7, c = idx & (HH - 1);
    const float* xr = x + n * FIN;
    float acc = 0.f;
#pragma unroll
    for (int k = 0; k < FIN; ++k) acc += xr[k] * W[k * HH + c];
    out[idx] = acc;
}

// BT[n*K + k] = B[k*ldn + n]  (weights are tiny; done once per GEMM)
__global__ void transpose_kernel(const float* __restrict__ B, float* __restrict__ BT,
                                 int K, int ldn) {
    int idx = blockIdx.x * 256 + threadIdx.x;   // exactly K*ldn threads
    int k = idx / ldn, n = idx % ldn;
    BT[(size_t)n * K + k] = B[idx];
}

// ------------------------------------------------------------ WMMA f32 GEMM
// D[M,ldn] = A[M,K] @ B[K,ldn] with B given transposed (BT[ldn,K]).
// Grid: (M/16, ldn/32); block: 32 threads = one wave per 16x32 strip.
// V_WMMA_F32_16X16X4_F32 VGPR layouts (ISA 7.12.2):
//   A 16x4:  lanes 0-15 -> M=lane, {K0,K1}; lanes 16-31 -> M=lane-16, {K2,K3}
//   B 4x16:  lanes 0-15 -> N=lane, {K0,K1}; lanes 16-31 -> N=lane-16, {K2,K3}
//   C/D:     VGPR v -> M = v + 8*(lane>=16), N = lane&15
__global__ void wmma_gemm_f32(const float* __restrict__ A, const float* __restrict__ BT,
                              const float* __restrict__ bias, float* __restrict__ D,
                              int K, int ldn, int relu) {
    int lane = threadIdx.x;
    int half = lane >> 4;          // 0 or 1
    int l    = lane & 15;
    const float* Ar  = A + (size_t)(blockIdx.x * 16 + l) * K;
    int col0 = blockIdx.y * 32 + l;
    int col1 = col0 + 16;
    const float* Bt0 = BT + (size_t)col0 * K;
    const float* Bt1 = BT + (size_t)col1 * K;

    v8f c0 = {}, c1 = {};
#pragma unroll 4
    for (int k0 = 0; k0 < K; k0 += 4) {
        int ka = k0 + 2 * half;                 // even -> 8B aligned
        v2f a  = *(const v2f*)(Ar + ka);
        v2f b0 = *(const v2f*)(Bt0 + ka);
        v2f b1 = *(const v2f*)(Bt1 + ka);
        c0 = __builtin_amdgcn_wmma_f32_16x16x4_f32(false, a, false, b0, (short)0, c0,
                                                   false, false);
        c1 = __builtin_amdgcn_wmma_f32_16x16x4_f32(false, a, false, b1, (short)0, c1,
                                                   false, false);
    }

    float bc0 = bias ? bias[col0] : 0.0f;
    float bc1 = bias ? bias[col1] : 0.0f;
    int rowb = blockIdx.x * 16 + 8 * half;
#pragma unroll
    for (int v = 0; v < 8; ++v) {
        float* Dr = D + (size_t)(rowb + v) * ldn;
        float v0 = c0[v] + bc0;
        float v1 = c1[v] + bc1;
        if (relu) { v0 = fmaxf(v0, 0.0f); v1 = fmaxf(v1, 0.0f); }
        Dr[col0] = v0;
        Dr[col1] = v1;
    }
}

// --------------------------------------------- GCN aggregation (CSR gather)
// One wave per node: lane handles 4 channels (32 lanes x 4 = 128).
// agg[n,:] = hW[n,:]*dis[n]^2 + bias + sum_{e: dst=n} hW[src_e,:]*dis[src]*dis[n]
__global__ void gather_agg_kernel(const float* __restrict__ hW, const int* __restrict__ src,
                                  const int* __restrict__ eidx, const int* __restrict__ rowstart,
                                  const float* __restrict__ dis, const float* __restrict__ b,
                                  float* __restrict__ agg) {
    int tid  = blockIdx.x * 256 + threadIdx.x;  // exactly N*32 threads
    int n    = tid >> 5;
    int c4   = (tid & 31) * 4;
    float dn = dis[n];

    float4 acc = *(const float4*)(hW + (size_t)n * HH + c4);
    const float4 bb = *(const float4*)(b + c4);
    float d2 = dn * dn;
    acc.x = acc.x * d2 + bb.x;
    acc.y = acc.y * d2 + bb.y;
    acc.z = acc.z * d2 + bb.z;
    acc.w = acc.w * d2 + bb.w;

    int e0 = rowstart[n], e1 = rowstart[n + 1];
    if (e0 < e1) {
        int s = src[eidx[e0]];
        for (int e = e0; e < e1; ++e) {
            int snext = (e + 1 < e1) ? src[eidx[e + 1]] : s;
            __builtin_prefetch(hW + (size_t)snext * HH + c4, 0, 1);
            float coef = dis[s] * dn;
            float4 h = *(const float4*)(hW + (size_t)s * HH + c4);
            acc.x += h.x * coef;
            acc.y += h.y * coef;
            acc.z += h.z * coef;
            acc.w += h.w * coef;
            s = snext;
        }
    }
    *(float4*)(agg + (size_t)n * HH + c4) = acc;
}

// -------------------------------------------------------------- batch norm
__global__ void bn_partial_kernel(const float* __restrict__ h, float* __restrict__ stats, int n) {
    int c  = threadIdx.x;                // 128 threads = channels
    int r0 = blockIdx.x * 128;
    int r1 = r0 + 128; if (r1 > n) r1 = n;
    float s = 0.f, s2 = 0.f;
    for (int r = r0; r < r1; ++r) {
        float v = h[(size_t)r * HH + c];
        s += v; s2 += v * v;
    }
    atomicAdd(&stats[c], s);
    atomicAdd(&stats[HH + c], s2);
}

__global__ void bn_finalize_kernel(const float* __restrict__ stats, const float* __restrict__ g,
                                   const float* __restrict__ be, float* __restrict__ out, int n) {
    int c = threadIdx.x;
    float inv = 1.0f / (float)n;
    float m = stats[c] * inv;
    float var = stats[HH + c] * inv - m * m;
    float sc = g[c] * rsqrtf(var + BN_EPS);
    out[c] = sc;
    out[HH + c] = be[c] - m * sc;
}

__global__ void bn_apply_kernel(float* __restrict__ h, const float* __restrict__ ss) {
    int idx = blockIdx.x * 256 + threadIdx.x;   // N*32 threads
    int c4 = (idx & 31) * 4;
    float4 sc = *(const float4*)(ss + c4);
    float4 sh = *(const float4*)(ss + HH + c4);
    float4 v = *(const float4*)(h + (size_t)idx * 4);
    v.x = fmaxf(v.x * sc.x + sh.x, 0.0f);
    v.y = fmaxf(v.y * sc.y + sh.y, 0.0f);
    v.z = fmaxf(v.z * sc.z + sh.z, 0.0f);
    v.w = fmaxf(v.w * sc.w + sh.w, 0.0f);
    *(float4*)(h + (size_t)idx * 4) = v;
}

// ------------------------------------------------------------------ pooling
// post-ReLU values are >= 0, so int atomicMax on the bit pattern is exact.
__global__ void pool_node_kernel(const float* __restrict__ h, const int* __restrict__ batch,
                                 float* __restrict__ psum, float* __restrict__ pmax,
                                 float* __restrict__ pcnt) {
    int idx = blockIdx.x * 256 + threadIdx.x;   // N*32 threads
    int n = idx >> 5, c4 = (idx & 31) * 4;
    int g = batch[n];
    float4 v = *(const float4*)(h + (size_t)n * HH + c4);
    float* sp = psum + (size_t)g * HH + c4;
    int*   mp = (int*)(pmax + (size_t)g * HH + c4);
    atomicAdd(sp + 0, v.x);
    atomicAdd(sp + 1, v.y);
    atomicAdd(sp + 2, v.z);
    atomicAdd(sp + 3, v.w);
    atomicMax(mp + 0, __float_as_int(v.x));
    atomicMax(mp + 1, __float_as_int(v.y));
    atomicMax(mp + 2, __float_as_int(v.z));
    atomicMax(mp + 3, __float_as_int(v.w));
    if (c4 == 0) atomicAdd(&pcnt[g], 1.0f);
}

__global__ void pool_final_kernel(const float* __restrict__ psum, const float* __restrict__ pmax,
                                  const float* __restrict__ pcnt, float* __restrict__ z) {
    int idx = blockIdx.x * 256 + threadIdx.x;   // G*H threads
    int g = idx >> 7, c = idx & (HH - 1);
    float cnt = fmaxf(pcnt[g], 1.0f);
    z[(size_t)g * (2 * HH) + c]      = psum[idx] / cnt;
    z[(size_t)g * (2 * HH) + HH + c] = pmax[idx];
}

// ---------------------------------------------------------------- out head
__global__ void out_kernel(const float* __restrict__ z2, const float* __restrict__ W,
                           const float* __restrict__ b, float* __restrict__ out) {
    int idx = blockIdx.x * blockDim.x + threadIdx.x;   // G*5
    if (idx >= GG * 5) return;
    int g = idx / 5, j = idx % 5;
    const float* zr = z2 + (size_t)g * 64;
    float acc = b[j];
#pragma unroll
    for (int k = 0; k < 64; ++k) acc += zr[k] * W[k * 5 + j];
    out[idx] = acc;
}

// ---------------------------------------------------------------------------
extern "C" void kernel_launch(void* const* d_in, const int* in_sizes, int n_in,
                              void* d_out, int out_size, void* d_ws, size_t ws_size,
                              hipStream_t stream) {
    const float* x     = (const float*)d_in[0];
    const int*   edges = (const int*)d_in[1];       // [2,E]: row0=src, row1=dst
    const int*   batch = (const int*)d_in[2];
    const float* W0 = (const float*)d_in[3],  *b0 = (const float*)d_in[4];
    const float* g0 = (const float*)d_in[5],  *be0 = (const float*)d_in[6];
    const float* W1 = (const float*)d_in[7],  *b1 = (const float*)d_in[8];
    const float* g1 = (const float*)d_in[9],  *be1 = (const float*)d_in[10];
    const float* W2 = (const float*)d_in[11], *b2 = (const float*)d_in[12];
    const float* g2 = (const float*)d_in[13], *be2 = (const float*)d_in[14];
    const float* fc1_w = (const float*)d_in[15], *fc1_b = (const float*)d_in[16];
    const float* fc2_w = (const float*)d_in[17], *fc2_b = (const float*)d_in[18];
    const float* out_w = (const float*)d_in[19], *out_b = (const float*)d_in[20];
    float* yout = (float*)d_out;

    const int* src = edges;
    const int* dst = edges + EE;

    // workspace carve-up (~60 MB, all regions 16B aligned)
    float* ws    = (float*)d_ws;
    float* dis   = ws;                       // N
    float* hW    = dis + 50176;              // N*H   (GEMM output / messages)
    float* hA    = hW  + (size_t)NN * HH;    // N*H   (aggregate / activations)
    float* stats = hA  + (size_t)NN * HH;    // 2*H
    float* scl   = stats + 2 * HH;           // 2*H
    float* psum  = scl + 2 * HH;             // G*H
    float* pmax  = psum + (size_t)GG * HH;   // G*H
    float* pcnt  = pmax + (size_t)GG * HH;   // G
    float* z     = pcnt + GG;                // G*2H
    float* z1    = z  + (size_t)GG * 2 * HH; // G*H
    float* z2    = z1 + (size_t)GG * HH;     // G*64
    float* wtr   = z2 + (size_t)GG * 64;     // 256*128 (transposed weights)
    int* rowstart = (int*)(wtr + 256 * HH);  // N+1 (padded to 50176)
    int* cursor   = rowstart + 50176;        // N
    int* eidx     = cursor + 50176;          // E

    const int NH_BLK  = (NN * HH) / 256;     // 25000  (1 float / thread)
    const int NV4_BLK = (NN * 32) / 256;     // 6250   (float4/thread or wave/node)
    const int E_BLK   = (EE + 255) / 256;

    // ---- CSR build: counts -> exclusive scan -> dis -> bucket scatter
    fill_i32<<<(NN + 255) / 256, 256, 0, stream>>>(rowstart, 0, NN);
    hist_kernel<<<E_BLK, 256, 0, stream>>>(dst, rowstart, EE);
    scan_kernel<<<1, 1024, 0, stream>>>(rowstart, NN);
    dis_kernel<<<(NN + 255) / 256, 256, 0, stream>>>(rowstart, dis, NN);
    copy_i32<<<(NN + 255) / 256, 256, 0, stream>>>(rowstart, cursor, NN);
    scatter_kernel<<<E_BLK, 256, 0, stream>>>(dst, cursor, eidx, EE);

    // one GCN+BN+ReLU layer; input activations live in hA afterwards
    auto gcn_bn = [&](const float* bias, const float* gam, const float* bet) {
        gather_agg_kernel<<<NV4_BLK, 256, 0, stream>>>(hW, src, eidx, rowstart, dis, bias, hA);
        fill_f32<<<1, 256, 0, stream>>>(stats, 0.0f, 2 * HH);
        bn_partial_kernel<<<(NN + 127) / 128, 128, 0, stream>>>(hA, stats, NN);
        bn_finalize_kernel<<<1, 128, 0, stream>>>(stats, gam, bet, scl, NN);
        bn_apply_kernel<<<NV4_BLK, 256, 0, stream>>>(hA, scl);
    };

    // ---- layer 0 (K=9 scalar GEMM)
    gemm0_kernel<<<NH_BLK, 256, 0, stream>>>(x, W0, hW);
    gcn_bn(b0, g0, be0);

    // ---- layer 1 (WMMA f32 GEMM: 50000x128x128)
    transpose_kernel<<<(HH * HH) / 256, 256, 0, stream>>>(W1, wtr, HH, HH);
    wmma_gemm_f32<<<dim3(NN / 16, HH / 32), 32, 0, stream>>>(hA, wtr, nullptr, hW, HH, HH, 0);
    gcn_bn(b1, g1, be1);

    // ---- layer 2
    transpose_kernel<<<(HH * HH) / 256, 256, 0, stream>>>(W2, wtr, HH, HH);
    wmma_gemm_f32<<<dim3(NN / 16, HH / 32), 32, 0, stream>>>(hA, wtr, nullptr, hW, HH, HH, 0);
    gcn_bn(b2, g2, be2);

    // ---- pooling (mean + max per graph)
    int poolN = GG * HH * 2 + GG;
    fill_f32<<<(poolN + 255) / 256, 256, 0, stream>>>(psum, 0.0f, poolN);
    pool_node_kernel<<<NV4_BLK, 256, 0, stream>>>(hA, batch, psum, pmax, pcnt);
    pool_final_kernel<<<(GG * HH) / 256, 256, 0, stream>>>(psum, pmax, pcnt, z);

    // ---- MLP head (WMMA): [G,256]@[256,128] relu, [G,128]@[128,64] relu
    transpose_kernel<<<(2 * HH * HH) / 256, 256, 0, stream>>>(fc1_w, wtr, 2 * HH, HH);
    wmma_gemm_f32<<<dim3(GG / 16, HH / 32), 32, 0, stream>>>(z, wtr, fc1_b, z1, 2 * HH, HH, 1);
    transpose_kernel<<<(HH * 64) / 256, 256, 0, stream>>>(fc2_w, wtr, HH, 64);
    wmma_gemm_f32<<<dim3(GG / 16, 64 / 32), 32, 0, stream>>>(z1, wtr, fc2_b, z2, HH, 64, 1);

    // ---- final [G,64]@[64,5]
    out_kernel<<<(GG * 5 + 255) / 256, 256, 0, stream>>>(z2, out_w, out_b, yout);
}